// Net_83124797047347
// MI455X (gfx1250) — compile-verified
//
#include <hip/hip_runtime.h>

typedef float v2f __attribute__((ext_vector_type(2)));
typedef float v8f __attribute__((ext_vector_type(8)));

#define N_NODES 100000
#define D_FEAT  128
#define HIDDEN  16

__device__ __forceinline__ void atomic_add_f32(float* p, float v) {
    // hardware global_atomic_add_f32 (no CAS loop)
    unsafeAtomicAdd(p, v);
}

// ---- init: deg = 1 (self loop), zero both aggregation buffers ----
__global__ void init_kernel(float* __restrict__ deg, float* __restrict__ agg1,
                            float* __restrict__ agg2, int n16, int n) {
    int i = blockIdx.x * blockDim.x + threadIdx.x;
    if (i < n)   deg[i]  = 1.0f;
    if (i < n16) { agg1[i] = 0.0f; agg2[i] = 0.0f; }
}

// ---- per-edge degree count ----
__global__ void deg_kernel(const int* __restrict__ ei, float* __restrict__ deg, int E) {
    int e = blockIdx.x * blockDim.x + threadIdx.x;
    if (e < E) atomic_add_f32(&deg[ei[E + e]], 1.0f);
}

// ---- dinv = rsqrt(deg) in place (deg >= 1 always) ----
__global__ void rsqrt_kernel(float* __restrict__ deg, int n) {
    int i = blockIdx.x * blockDim.x + threadIdx.x;
    if (i < n) deg[i] = rsqrtf(deg[i]);
}

// ---- GEMM1: h1[N x 16] = X[N x 128] @ W1[128 x 16], f32 WMMA 16x16x4 ----
__global__ __launch_bounds__(256) void gemm1_wmma(const float* __restrict__ x,
                                                  const float* __restrict__ W1,
                                                  float* __restrict__ h1, int nTiles) {
    __shared__ float sW[D_FEAT * HIDDEN]; // 8 KB
    for (int i = threadIdx.x; i < D_FEAT * HIDDEN; i += blockDim.x) sW[i] = W1[i];
    __syncthreads();

    int wid = blockIdx.x * (blockDim.x >> 5) + (threadIdx.x >> 5);
    if (wid >= nTiles) return;          // wave-uniform; EXEC all-1 inside

    int lane = threadIdx.x & 31;
    int lh   = lane >> 4;               // 0: K pair {0,1}; 1: K pair {2,3}
    int lm   = lane & 15;               // M for A, N for B/C
    int m0   = wid * 16;

    v8f c = {};
    const float* xrow = x + (size_t)(m0 + lm) * D_FEAT;
#pragma unroll
    for (int k = 0; k < D_FEAT / 4; ++k) {
        int kk = k * 4 + 2 * lh;
        v2f a, b;
        a.x = xrow[kk];
        a.y = xrow[kk + 1];
        b.x = sW[kk * HIDDEN + lm];
        b.y = sW[(kk + 1) * HIDDEN + lm];
        c = __builtin_amdgcn_wmma_f32_16x16x4_f32(false, a, false, b,
                                                  (short)0, c, false, false);
    }
    float* hrow = h1 + (size_t)m0 * HIDDEN;
#pragma unroll
    for (int i = 0; i < 8; ++i)
        hrow[(i + 8 * lh) * HIDDEN + lm] = c[i];
}

// ---- edge aggregation in 16-dim space: agg[d] += dinv[s]*dinv[d]*h[s] ----
// 4 lanes per edge, each handling a float4 slice; f32 hardware atomics.
__global__ void agg_kernel(const int* __restrict__ ei, const float* __restrict__ dinv,
                           const float* __restrict__ h, float* __restrict__ agg, int E) {
    int gid = blockIdx.x * blockDim.x + threadIdx.x;
    int e = gid >> 2;
    if (e >= E) return;
    int p = (gid & 3) * 4;
    int s = ei[e];
    int d = ei[E + e];
    float w = dinv[s] * dinv[d];
    const float4 v = *(const float4*)(h + (size_t)s * HIDDEN + p);
    float* o = agg + (size_t)d * HIDDEN + p;
    atomic_add_f32(o + 0, v.x * w);
    atomic_add_f32(o + 1, v.y * w);
    atomic_add_f32(o + 2, v.z * w);
    atomic_add_f32(o + 3, v.w * w);
}

// ---- layer-1 epilogue: self-loop + bias + ReLU, in place ----
__global__ void selfloop_relu_kernel(float* __restrict__ agg1, const float* __restrict__ h1,
                                     const float* __restrict__ dinv,
                                     const float* __restrict__ b1, int n16) {
    int gid = blockIdx.x * blockDim.x + threadIdx.x;
    if (gid >= n16) return;
    int nId = gid >> 4, j = gid & 15;
    float di = dinv[nId];
    float v = agg1[gid] + di * di * h1[gid] + b1[j];
    agg1[gid] = v > 0.0f ? v : 0.0f;
}

// ---- layer-2 self-loop add, in place ----
__global__ void selfloop2_kernel(float* __restrict__ agg2, const float* __restrict__ hbuf,
                                 const float* __restrict__ dinv, int n16) {
    int gid = blockIdx.x * blockDim.x + threadIdx.x;
    if (gid >= n16) return;
    float di = dinv[gid >> 4];
    agg2[gid] += di * di * hbuf[gid];
}

// ---- GEMM2: out[N x 128] = A[N x 16] @ W2[16 x 128] + b2, f32 WMMA ----
// One wave per 16-node tile, 8 accumulators covering all 128 output columns.
__global__ __launch_bounds__(256) void gemm2_wmma(const float* __restrict__ A,
                                                  const float* __restrict__ W2,
                                                  const float* __restrict__ b2,
                                                  float* __restrict__ out, int nTiles) {
    __shared__ float sW[HIDDEN * D_FEAT]; // 8 KB
    for (int i = threadIdx.x; i < HIDDEN * D_FEAT; i += blockDim.x) sW[i] = W2[i];
    __syncthreads();

    int wid = blockIdx.x * (blockDim.x >> 5) + (threadIdx.x >> 5);
    if (wid >= nTiles) return;

    int lane = threadIdx.x & 31;
    int lh   = lane >> 4;
    int lm   = lane & 15;
    int m0   = wid * 16;

    v8f c[8];
#pragma unroll
    for (int t = 0; t < 8; ++t) c[t] = (v8f){};

    const float* arow = A + (size_t)(m0 + lm) * HIDDEN;
#pragma unroll
    for (int k = 0; k < HIDDEN / 4; ++k) {
        int kk = k * 4 + 2 * lh;
        v2f a;
        a.x = arow[kk];
        a.y = arow[kk + 1];
#pragma unroll
        for (int t = 0; t < 8; ++t) {
            v2f b;
            b.x = sW[kk * D_FEAT + t * 16 + lm];
            b.y = sW[(kk + 1) * D_FEAT + t * 16 + lm];
            c[t] = __builtin_amdgcn_wmma_f32_16x16x4_f32(false, a, false, b,
                                                         (short)0, c[t], false, false);
        }
    }
#pragma unroll
    for (int t = 0; t < 8; ++t) {
        float bias = b2[t * 16 + lm];
        float* orow = out + (size_t)m0 * D_FEAT + t * 16 + lm;
#pragma unroll
        for (int i = 0; i < 8; ++i)
            orow[(size_t)(i + 8 * lh) * D_FEAT] = c[t][i] + bias;
    }
}

extern "C" void kernel_launch(void* const* d_in, const int* in_sizes, int n_in,
                              void* d_out, int out_size, void* d_ws, size_t ws_size,
                              hipStream_t stream) {
    const float* x  = (const float*)d_in[0];
    const int*   ei = (const int*)d_in[1];   // edge_index [2, E]
    const float* W1 = (const float*)d_in[2];
    const float* b1 = (const float*)d_in[3];
    const float* W2 = (const float*)d_in[4];
    const float* b2 = (const float*)d_in[5];
    float* out = (float*)d_out;

    const int E   = in_sizes[1] / 2;
    const int n   = N_NODES;
    const int n16 = n * HIDDEN;

    float* ws   = (float*)d_ws;
    float* dinv = ws;              // n floats (holds deg, then rsqrt in place)
    float* h1   = ws + 100352;     // n16 floats (X @ W1)
    float* agg1 = h1 + n16;        // n16 floats (layer-1 agg -> relu'd hidden)
    float* agg2 = agg1 + n16;      // n16 floats (layer-2 agg)

    const int nTiles = (n + 15) / 16;   // 6250 (exact)
    const int TB = 256;

    init_kernel<<<(n16 + TB - 1) / TB, TB, 0, stream>>>(dinv, agg1, agg2, n16, n);
    deg_kernel<<<(E + TB - 1) / TB, TB, 0, stream>>>(ei, dinv, E);
    rsqrt_kernel<<<(n + TB - 1) / TB, TB, 0, stream>>>(dinv, n);
    gemm1_wmma<<<(nTiles + 7) / 8, TB, 0, stream>>>(x, W1, h1, nTiles);
    agg_kernel<<<(E * 4 + TB - 1) / TB, TB, 0, stream>>>(ei, dinv, h1, agg1, E);
    selfloop_relu_kernel<<<(n16 + TB - 1) / TB, TB, 0, stream>>>(agg1, h1, dinv, b1, n16);
    agg_kernel<<<(E * 4 + TB - 1) / TB, TB, 0, stream>>>(ei, dinv, agg1, agg2, E);
    selfloop2_kernel<<<(n16 + TB - 1) / TB, TB, 0, stream>>>(agg2, agg1, dinv, n16);
    gemm2_wmma<<<(nTiles + 7) / 8, TB, 0, stream>>>(agg2, W2, b2, out, nTiles);
}